// SoftGumbelQuantizer_7645041786973
// MI455X (gfx1250) — compile-verified
//
#include <hip/hip_runtime.h>
#include <hip/hip_bf16.h>
#include <stdint.h>

// Problem constants (from reference): N=32, C=512, T=2048
#define NSEQ  32
#define CDIM  512
#define TDIM  2048
#define TTILE 32                 // t-values per block
#define XROW  33                 // padded LDS row stride (floats) for x tile (TDM pad: 32 dw + 1 dw)
#define WROW  513                // padded LDS row stride (floats) for W rows
#define NTROWS (NSEQ * TDIM)     // 65536 rows

typedef unsigned int v4u __attribute__((ext_vector_type(4)));
typedef int          v8i __attribute__((ext_vector_type(8)));
typedef int          v4i __attribute__((ext_vector_type(4)));

#if defined(__gfx1250__) && __has_builtin(__builtin_amdgcn_tensor_load_to_lds)
#define HAVE_TDM 1
#else
#define HAVE_TDM 0
#endif

#if HAVE_TDM
// Issue a TDM 2D tile load: 512 rows (c, stride T floats) x 32 floats (t),
// into LDS at lds_off with 1-DWORD pad after every 32 DWORDs (row stride 33).
__device__ __forceinline__ void tdm_load_tile(const float* gsrc, unsigned lds_off) {
  unsigned long long ga = (unsigned long long)(uintptr_t)gsrc;
  v4u g0;
  g0.x = 1u;                                                  // count=1 valid descriptor
  g0.y = lds_off;                                             // lds_addr (bytes)
  g0.z = (unsigned)(ga & 0xFFFFFFFFull);                      // global_addr[31:0]
  g0.w = (unsigned)((ga >> 32) & 0x01FFFFFFull) | (2u << 30); // global_addr[56:32], type=2
  v8i g1;
  // data_size=2 (4B), pad_enable=1, pad_interval=4 (32 DWORDs), pad_amount=0 (1 DWORD)
  g1[0] = (int)((2u << 16) | (1u << 20) | (4u << 22));
  g1[1] = (int)(((unsigned)TDIM & 0xFFFFu) << 16);            // tensor_dim0[15:0] (=2048)
  g1[2] = (int)((((unsigned)TDIM >> 16) & 0xFFFFu) |          // tensor_dim0[31:16]
                (((unsigned)CDIM & 0xFFFFu) << 16));          // tensor_dim1[15:0] (=512)
  g1[3] = (int)((((unsigned)CDIM >> 16) & 0xFFFFu) |          // tensor_dim1[31:16]
                ((unsigned)TTILE << 16));                     // tile_dim0 = 32
  g1[4] = (int)CDIM;                                          // tile_dim1=512, tile_dim2=0
  g1[5] = (int)TDIM;                                          // tensor_dim0_stride low32 = 2048
  g1[6] = 0;                                                  // stride0 hi / stride1 lo (unused)
  g1[7] = 0;
  v4i z4 = (v4i){0, 0, 0, 0};
#if defined(__clang_major__) && (__clang_major__ >= 23)
  v8i z8 = (v8i){0, 0, 0, 0, 0, 0, 0, 0};
  __builtin_amdgcn_tensor_load_to_lds(g0, g1, z4, z4, z8, 0);
#else
  __builtin_amdgcn_tensor_load_to_lds(g0, g1, z4, z4, 0);
#endif
}
#endif

__device__ __forceinline__ float gumbel_noise(float u) {
  u = fminf(fmaxf(u, 1e-10f), 1.0f - 1e-7f);
  return -__logf(-__logf(u));
}

// Fused: x-tile (TDM) -> LDS transpose -> per-row argmax(x + gumbel(u2))
//        -> histogram atomic -> gather W rows via LDS -> coalesced transposed store.
__global__ __launch_bounds__(256) void sgq_main(const float* __restrict__ xenc,
                                                const float* __restrict__ u2,
                                                const float* __restrict__ W,
                                                float* __restrict__ out,
                                                unsigned int* __restrict__ hist) {
  __shared__ float smem[CDIM * XROW];   // 67584 B; re-used for W rows in phase B
  __shared__ int   s_idx[TTILE];

  const int tid   = threadIdx.x;
  const int lane  = tid & 31;
  const int wave  = tid >> 5;
  const int blk   = blockIdx.x;
  const int n     = blk >> 6;           // 64 t-tiles per n
  const int tbase = (blk & 63) << 5;

  const float* tile_src = xenc + ((size_t)n * CDIM) * TDIM + tbase;

#if HAVE_TDM
  if (wave == 0) {
    tdm_load_tile(tile_src, (unsigned)(uintptr_t)(&smem[0]));
#if __has_builtin(__builtin_amdgcn_s_wait_tensorcnt)
    __builtin_amdgcn_s_wait_tensorcnt(0);
#else
    asm volatile("s_wait_tensorcnt 0" ::: "memory");
#endif
  }
#else
  // Fallback: coalesced 128-bit loads, LDS layout [c][t] with stride XROW
  for (int e = tid; e < CDIM * (TTILE / 4); e += 256) {
    int c = e >> 3;
    int q = (e & 7) << 2;
    float4 v = *(const float4*)(tile_src + (size_t)c * TDIM + q);
    float* dst = &smem[c * XROW + q];
    dst[0] = v.x; dst[1] = v.y; dst[2] = v.z; dst[3] = v.w;
  }
#endif
  __syncthreads();

  // ---- Phase A: argmax over C per row; 4 rows per wave ----
  #pragma unroll
  for (int rr = 0; rr < 4; ++rr) {
    const int r = wave * 4 + rr;
    const size_t rowi = (size_t)n * TDIM + tbase + r;
    const float* urow = u2 + rowi * CDIM;
    if (rr < 3) __builtin_prefetch(urow + CDIM, 0, 0);  // next row of this wave

    float best = -INFINITY;
    int bestc = 0;
    #pragma unroll
    for (int k = 0; k < CDIM / 32; ++k) {
      const int c = lane + (k << 5);
      const float v = smem[c * XROW + r] + gumbel_noise(urow[c]);
      if (v > best) { best = v; bestc = c; }   // ascending c => keeps lowest on ties
    }
    // wave32 argmax reduction, lowest-index tie-break (matches jnp.argmax)
    #pragma unroll
    for (int off = 16; off > 0; off >>= 1) {
      const float ov = __shfl_xor(best, off, 32);
      const int   oc = __shfl_xor(bestc, off, 32);
      if (ov > best || (ov == best && oc < bestc)) { best = ov; bestc = oc; }
    }
    if (lane == 0) {
      s_idx[r] = bestc;
      atomicAdd(&hist[bestc], 1u);
    }
  }
  __syncthreads();

  // ---- Phase B: stage the 32 selected W rows in LDS (padded), then scatter-free writes ----
  for (int e = tid; e < TTILE * (CDIM / 4); e += 256) {
    const int r = e >> 7;
    const int d = (e & 127) << 2;
    const float4 w4 = *(const float4*)(W + (size_t)s_idx[r] * CDIM + d);
    float* dst = &smem[r * WROW + d];
    dst[0] = w4.x; dst[1] = w4.y; dst[2] = w4.z; dst[3] = w4.w;
  }
  __syncthreads();

  // out[(n*512 + d)*T + tbase + lane]; lane sweeps t => coalesced 128B stores per wave.
  float* obase = out + ((size_t)n * CDIM) * TDIM + tbase + lane;
  #pragma unroll 4
  for (int k = 0; k < CDIM / 8; ++k) {
    const int d = wave + (k << 3);
    obase[(size_t)d * TDIM] = smem[lane * WROW + d];  // (lane + d) % 64 banks: conflict-free
  }
}

// Histogram -> perplexity, loss_util; loss_self_entropy is exactly 0 (ENT_COEF==0).
__global__ __launch_bounds__(512) void sgq_finalize(const unsigned int* __restrict__ hist,
                                                    float* __restrict__ scal) {
  __shared__ float red[CDIM];
  const int c = threadIdx.x;
  const float p = (float)hist[c] * (1.0f / (float)NTROWS);
  red[c] = -p * logf(p + 1e-7f);
  __syncthreads();
  for (int s = 256; s > 0; s >>= 1) {
    if (c < s) red[c] += red[c + s];
    __syncthreads();
  }
  if (c == 0) {
    const float perp = expf(red[0]);
    scal[0] = (1.0f - perp * (1.0f / 512.0f)) * 5.0f;  // loss_util (RATIO=5, HARD_PPL_RATE=0)
    scal[1] = 0.0f;                                    // loss_self_entropy (ENT_COEF=0)
    scal[2] = perp;                                    // perplexity
  }
}

extern "C" void kernel_launch(void* const* d_in, const int* in_sizes, int n_in,
                              void* d_out, int out_size, void* d_ws, size_t ws_size,
                              hipStream_t stream) {
  const float* xenc = (const float*)d_in[0];
  // d_in[1] = u1 : dead (only feeds loss_self_entropy * ENT_COEF, ENT_COEF == 0)
  const float* u2 = (const float*)d_in[2];
  const float* W  = (const float*)d_in[3];
  float* out = (float*)d_out;
  unsigned int* hist = (unsigned int*)d_ws;

  hipMemsetAsync(hist, 0, CDIM * sizeof(unsigned int), stream);  // deterministic per call

  const int nblocks = NSEQ * (TDIM / TTILE);  // 2048
  sgq_main<<<nblocks, 256, 0, stream>>>(xenc, u2, W, out, hist);
  sgq_finalize<<<1, CDIM, 0, stream>>>(hist, out + (size_t)NSEQ * CDIM * TDIM);
}